// SlidingWindowCache_45861660787372
// MI455X (gfx1250) — compile-verified
//
#include <hip/hip_runtime.h>

// Sliding-window KV cache "gather" — reduces to a contiguous slice copy:
//   out_k[b, i, :, :] = key_t  [b, i + 512, :, :],  i in [0, 4096)
//   out_v[b, i, :, :] = value_t[b, i + 512, :, :]
// Pure HBM-bandwidth problem: 512 MiB total traffic -> ~22 us at 23.3 TB/s.
// Strategy: b128 (float4) coalesced loads/stores with non-temporal hints
// (working set 512 MiB >> 192 MB L2; every byte touched once).

typedef float v4f __attribute__((ext_vector_type(4)));

constexpr long long B = 8;
constexpr long long S = 4608;
constexpr long long H = 8;
constexpr long long D = 128;
constexpr long long W = 4096;
constexpr long long START = S % W;                    // 512
constexpr long long ROW = H * D;                      // 1024 floats / token
constexpr long long SRC_BATCH4 = S * ROW / 4;         // float4 per src batch = 1,179,648
constexpr long long DST_BATCH4 = W * ROW / 4;         // float4 per dst batch = 1,048,576 (2^20)
constexpr long long SRC_OFF4   = START * ROW / 4;     // slice offset in float4 = 131,072
constexpr long long N4_TOTAL   = B * DST_BATCH4;      // 8,388,608 float4 per tensor

__global__ __launch_bounds__(256)
void SlidingWindowCache_copy_kernel(const v4f* __restrict__ k,
                                    const v4f* __restrict__ v,
                                    v4f* __restrict__ ok,
                                    v4f* __restrict__ ov) {
    long long i = (long long)blockIdx.x * blockDim.x + threadIdx.x;
    if (i >= N4_TOTAL) return;

    long long b = i >> 20;                 // i / DST_BATCH4 (2^20)
    long long r = i & (DST_BATCH4 - 1);    // i % DST_BATCH4
    long long s = b * SRC_BATCH4 + SRC_OFF4 + r;

    v4f kq = __builtin_nontemporal_load(&k[s]);
    v4f vq = __builtin_nontemporal_load(&v[s]);
    __builtin_nontemporal_store(kq, &ok[i]);
    __builtin_nontemporal_store(vq, &ov[i]);
}

extern "C" void kernel_launch(void* const* d_in, const int* in_sizes, int n_in,
                              void* d_out, int out_size, void* d_ws, size_t ws_size,
                              hipStream_t stream) {
    (void)in_sizes; (void)n_in; (void)out_size; (void)d_ws; (void)ws_size;

    const v4f* k = (const v4f*)d_in[0];
    const v4f* v = (const v4f*)d_in[1];

    float* out = (float*)d_out;
    v4f* ok = (v4f*)out;                                   // cached_k: B*W*H*D floats
    v4f* ov = (v4f*)(out + B * W * H * D);                 // cached_v follows

    const int block = 256;                                 // 8 wave32 waves
    const long long grid = (N4_TOTAL + block - 1) / block; // 32768 blocks

    SlidingWindowCache_copy_kernel<<<(unsigned)grid, block, 0, stream>>>(k, v, ok, ov);
}